// find_NN_batch_51230369907085
// MI455X (gfx1250) — compile-verified
//
#include <hip/hip_runtime.h>

typedef unsigned int u32;
typedef unsigned long long u64;

#define BATCH 32
#define NPTS  262144
#define KSEL  1024
#define BINS  4096
#define TILE  2048
#define TPB   256
#define TILES_PER_BATCH (NPTS / TILE)     // 128
#define TILES_PER_WG    8
#define WGS_PER_BATCH   (TILES_PER_BATCH / TILES_PER_WG)  // 16

#ifndef __has_builtin
#define __has_builtin(x) 0
#endif
#if __has_builtin(__builtin_amdgcn_tensor_load_to_lds)
#define HAVE_TDM 1
#else
#define HAVE_TDM 0
#endif

#if HAVE_TDM
typedef __attribute__((ext_vector_type(4))) unsigned int v4u;
typedef __attribute__((ext_vector_type(4))) int v4i;
typedef __attribute__((ext_vector_type(8))) int v8i;

// Build a Tensor DMA Descriptor (D#) per CDNA5 ISA ch.8 and issue
// TENSOR_LOAD_TO_LDS: a 2D tile (nRows x rowElems f32), rows rowStrideElems
// apart in global memory, packed contiguously into LDS.
__device__ __forceinline__ void tdm_load_rows(void* ldsDst, const void* gsrc,
                                              u32 rowElems, u32 nRows,
                                              u64 rowStrideElems) {
  u64 ga  = (u64)(size_t)gsrc;
  u32 lds = (u32)(u64)(size_t)ldsDst;   // low 32 bits of flat LDS aperture addr
  v4u g0;
  g0[0] = 1u;                                            // count=1, user D#
  g0[1] = lds;                                           // lds_addr
  g0[2] = (u32)ga;                                       // global_addr[31:0]
  g0[3] = ((u32)(ga >> 32) & 0x01FFFFFFu) | (2u << 30);  // addr[56:32] | type=2
  v8i g1;
  g1[0] = (int)(2u << 16);                               // data_size=4B, no mcast
  g1[1] = (int)(rowElems << 16);                         // tensor_dim0[15:0]
  g1[2] = (int)((rowElems >> 16) | (nRows << 16));       // dim0 hi | tensor_dim1 lo
  g1[3] = (int)((nRows >> 16) | (rowElems << 16));       // dim1 hi | tile_dim0
  g1[4] = (int)(nRows & 0xFFFFu);                        // tile_dim1 (tile_dim2=0)
  g1[5] = (int)(u32)(rowStrideElems & 0xFFFFFFFFull);    // dim0_stride[31:0]
  g1[6] = (int)(u32)((rowStrideElems >> 32) & 0xFFFFull);// dim0_stride[47:32]
  g1[7] = 0;
  v4i gz = {0, 0, 0, 0};
#if defined(__clang_major__) && (__clang_major__ >= 23)
  v8i gz8 = {0, 0, 0, 0, 0, 0, 0, 0};
  __builtin_amdgcn_tensor_load_to_lds(g0, g1, gz, gz, gz8, 0);
#else
  __builtin_amdgcn_tensor_load_to_lds(g0, g1, gz, gz, 0);
#endif
}
#endif  // HAVE_TDM

__device__ __forceinline__ u32 dist_key(const float* tile, int i,
                                        float p0, float p1, float p2) {
  float dx = tile[i] - p0;
  float dy = tile[TILE + i] - p1;
  float dz = tile[2 * TILE + i] - p2;
  float d2 = fmaf(dx, dx, fmaf(dy, dy, dz * dz));
  return __float_as_uint(d2);  // d2 >= 0 -> bit pattern is order-preserving
}

// Double-buffered pipeline helpers -----------------------------------------
// Issue DMA for tile (t0+tt+1) into buffer cur^1 (if any), then wait until the
// current buffer's DMA (older, in-order) has completed.
__device__ __forceinline__ void pipe_step(float (*tile)[3 * TILE],
                                          const float* base, int t0, int tt,
                                          int cur) {
#if HAVE_TDM
  if (threadIdx.x < 32) {
    if (tt + 1 < TILES_PER_WG) {
      tdm_load_rows(&tile[cur ^ 1][0], base + (size_t)(t0 + tt + 1) * TILE,
                    TILE, 3, (u64)NPTS);
#if __has_builtin(__builtin_amdgcn_s_wait_tensorcnt)
      __builtin_amdgcn_s_wait_tensorcnt(1);  // older (cur) DMA done, newer in flight
#endif
    } else {
#if __has_builtin(__builtin_amdgcn_s_wait_tensorcnt)
      __builtin_amdgcn_s_wait_tensorcnt(0);
#endif
    }
  }
#else
  for (int c = 0; c < 3; ++c)
    for (int i = threadIdx.x; i < TILE; i += TPB)
      tile[cur][c * TILE + i] =
          base[(size_t)c * NPTS + (size_t)(t0 + tt) * TILE + i];
#endif
  __syncthreads();  // DMA data visible to all waves
}

__device__ __forceinline__ void pipe_prologue(float (*tile)[3 * TILE],
                                              const float* base, int t0) {
#if HAVE_TDM
  if (threadIdx.x < 32)
    tdm_load_rows(&tile[0][0], base + (size_t)t0 * TILE, TILE, 3, (u64)NPTS);
#endif
}

__global__ void k_zero(u32* p, int n) {
  int i = blockIdx.x * blockDim.x + threadIdx.x;
  if (i < n) p[i] = 0u;
}

// Sweep 1: per-batch histogram of top-12 key bits (LDS hist, global flush).
__global__ __launch_bounds__(TPB) void k_hist1(const float* __restrict__ pcloud,
                                               const float* __restrict__ P1,
                                               u32* __restrict__ hist1) {
  __shared__ float tile[2][3 * TILE];
  __shared__ u32 lhist[BINS];
  const int b = blockIdx.x / WGS_PER_BATCH;
  const int t0 = (blockIdx.x % WGS_PER_BATCH) * TILES_PER_WG;
  for (int i = threadIdx.x; i < BINS; i += TPB) lhist[i] = 0u;
  const float* base = pcloud + (size_t)b * 3 * NPTS;
  pipe_prologue(tile, base, t0);
  const float p0 = P1[b * 3 + 0], p1 = P1[b * 3 + 1], p2 = P1[b * 3 + 2];
  for (int tt = 0; tt < TILES_PER_WG; ++tt) {
    const int cur = tt & 1;
    pipe_step(tile, base, t0, tt, cur);
    for (int i = threadIdx.x; i < TILE; i += TPB) {
      u32 bin = dist_key(&tile[cur][0], i, p0, p1, p2) >> 20;
      atomicAdd(&lhist[bin], 1u);
    }
    __syncthreads();  // buffer free before next-iteration DMA reuses it
  }
  for (int i = threadIdx.x; i < BINS; i += TPB) {
    u32 v = lhist[i];
    if (v) atomicAdd(&hist1[(size_t)b * BINS + i], v);
  }
}

__global__ void k_scan1(const u32* __restrict__ hist1, u32* thr1, u32* cb1) {
  if (threadIdx.x != 0) return;
  const int b = blockIdx.x;
  u32 c = 0, T = BINS - 1;
  for (int i = 0; i < BINS; ++i) {
    u32 h = hist1[(size_t)b * BINS + i];
    if (c + h >= KSEL) { T = (u32)i; break; }
    c += h;
  }
  thr1[b] = T;
  cb1[b] = c;
}

// Sweep 2: refine threshold bin with the next 12 key bits.
__global__ __launch_bounds__(TPB) void k_hist2(const float* __restrict__ pcloud,
                                               const float* __restrict__ P1,
                                               const u32* __restrict__ thr1,
                                               u32* __restrict__ hist2) {
  __shared__ float tile[2][3 * TILE];
  const int b = blockIdx.x / WGS_PER_BATCH;
  const int t0 = (blockIdx.x % WGS_PER_BATCH) * TILES_PER_WG;
  const float* base = pcloud + (size_t)b * 3 * NPTS;
  pipe_prologue(tile, base, t0);
  const float p0 = P1[b * 3 + 0], p1 = P1[b * 3 + 1], p2 = P1[b * 3 + 2];
  const u32 T1 = thr1[b];
  for (int tt = 0; tt < TILES_PER_WG; ++tt) {
    const int cur = tt & 1;
    pipe_step(tile, base, t0, tt, cur);
    for (int i = threadIdx.x; i < TILE; i += TPB) {
      u32 key = dist_key(&tile[cur][0], i, p0, p1, p2);
      if ((key >> 20) == T1)
        atomicAdd(&hist2[(size_t)b * BINS + ((key >> 8) & 0xFFFu)], 1u);
    }
    __syncthreads();
  }
}

__global__ void k_scan2(const u32* __restrict__ hist2, const u32* __restrict__ cb1,
                        u32* thr2, u32* cb2) {
  if (threadIdx.x != 0) return;
  const int b = blockIdx.x;
  const u32 need = KSEL - cb1[b];
  u32 c = 0, T = BINS - 1;
  for (int i = 0; i < BINS; ++i) {
    u32 h = hist2[(size_t)b * BINS + i];
    if (c + h >= need) { T = (u32)i; break; }
    c += h;
  }
  thr2[b] = T;
  cb2[b] = cb1[b] + c;  // total count strictly below refined 24-bit threshold
}

// Sweep 3: collect the K winners (definite + threshold-tie fill).
__global__ __launch_bounds__(TPB) void k_collect(
    const float* __restrict__ pcloud, const float* __restrict__ P1,
    const u32* __restrict__ thr1, const u32* __restrict__ thr2,
    const u32* __restrict__ cb2, u32* __restrict__ cntA, u32* __restrict__ cntB,
    u32* __restrict__ ckey, u32* __restrict__ cidx) {
  __shared__ float tile[2][3 * TILE];
  const int b = blockIdx.x / WGS_PER_BATCH;
  const int t0 = (blockIdx.x % WGS_PER_BATCH) * TILES_PER_WG;
  const float* base = pcloud + (size_t)b * 3 * NPTS;
  pipe_prologue(tile, base, t0);
  const float p0 = P1[b * 3 + 0], p1 = P1[b * 3 + 1], p2 = P1[b * 3 + 2];
  const u32 T1 = thr1[b], T2 = thr2[b], slotBase = cb2[b];
  for (int tt = 0; tt < TILES_PER_WG; ++tt) {
    const int cur = tt & 1;
    pipe_step(tile, base, t0, tt, cur);
    for (int i = threadIdx.x; i < TILE; i += TPB) {
      u32 key = dist_key(&tile[cur][0], i, p0, p1, p2);
      u32 b1 = key >> 20;
      if (b1 > T1) continue;
      u32 sub = (key >> 8) & 0xFFFu;
      u32 gidx = (u32)((t0 + tt) * TILE + i);
      if (b1 < T1 || sub < T2) {
        u32 pos = atomicAdd(&cntA[b], 1u);
        if (pos < KSEL) {
          ckey[(size_t)b * KSEL + pos] = key;
          cidx[(size_t)b * KSEL + pos] = gidx;
        }
      } else if (sub == T2) {
        u32 pos = atomicAdd(&cntB[b], 1u);
        u32 slot = slotBase + pos;
        if (slot < KSEL) {
          ckey[(size_t)b * KSEL + slot] = key;
          cidx[(size_t)b * KSEL + slot] = gidx;
        }
      }
    }
    __syncthreads();
  }
}

// Per-batch bitonic sort of K=1024 (key,idx) pairs, then gather outputs.
__global__ __launch_bounds__(KSEL) void k_sort_gather(
    const float* __restrict__ pcloud, const u32* __restrict__ ckey,
    const u32* __restrict__ cidx, float* __restrict__ out) {
  __shared__ u32 sk[KSEL];
  __shared__ u32 si[KSEL];
  const int b = blockIdx.x;
  const int tid = threadIdx.x;
  sk[tid] = ckey[(size_t)b * KSEL + tid];
  si[tid] = cidx[(size_t)b * KSEL + tid];
  __syncthreads();
  for (int k = 2; k <= KSEL; k <<= 1) {
    for (int j = k >> 1; j > 0; j >>= 1) {
      int ixj = tid ^ j;
      if (ixj > tid) {
        bool up = ((tid & k) == 0);
        u32 a = sk[tid], c = sk[ixj];
        if (up ? (a > c) : (a < c)) {
          u32 ia = si[tid], ic = si[ixj];
          sk[tid] = c; sk[ixj] = a;
          si[tid] = ic; si[ixj] = ia;
        }
      }
      __syncthreads();
    }
  }
  const u32 idx = si[tid];
  float* nearest = out;                             // (B, 3, K)
  float* oidx = out + (size_t)BATCH * 3 * KSEL;     // (B, K), written as float
  for (int c = 0; c < 3; ++c)
    nearest[((size_t)b * 3 + c) * KSEL + tid] =
        pcloud[((size_t)b * 3 + c) * NPTS + idx];
  oidx[(size_t)b * KSEL + tid] = (float)idx;
}

extern "C" void kernel_launch(void* const* d_in, const int* in_sizes, int n_in,
                              void* d_out, int out_size, void* d_ws, size_t ws_size,
                              hipStream_t stream) {
  (void)in_sizes; (void)n_in; (void)out_size; (void)ws_size;
  const float* pcloud = (const float*)d_in[0];
  const float* P1 = (const float*)d_in[1];
  float* out = (float*)d_out;

  u32* ws    = (u32*)d_ws;
  u32* hist1 = ws;                                // B*BINS
  u32* hist2 = hist1 + (size_t)BATCH * BINS;      // B*BINS
  u32* thr1  = hist2 + (size_t)BATCH * BINS;      // B
  u32* cb1   = thr1 + BATCH;                      // B
  u32* thr2  = cb1 + BATCH;                       // B
  u32* cb2   = thr2 + BATCH;                      // B
  u32* cntA  = cb2 + BATCH;                       // B
  u32* cntB  = cntA + BATCH;                      // B
  u32* ckey  = cntB + BATCH;                      // B*K
  u32* cidx  = ckey + (size_t)BATCH * KSEL;       // B*K

  const int zn = 2 * BATCH * BINS + 6 * BATCH;    // hist1+hist2+scalars
  k_zero<<<(zn + 255) / 256, 256, 0, stream>>>(ws, zn);

  const int wgs = BATCH * WGS_PER_BATCH;          // 512 pipelined workgroups
  k_hist1<<<wgs, TPB, 0, stream>>>(pcloud, P1, hist1);
  k_scan1<<<BATCH, 32, 0, stream>>>(hist1, thr1, cb1);
  k_hist2<<<wgs, TPB, 0, stream>>>(pcloud, P1, thr1, hist2);
  k_scan2<<<BATCH, 32, 0, stream>>>(hist2, cb1, thr2, cb2);
  k_collect<<<wgs, TPB, 0, stream>>>(pcloud, P1, thr1, thr2, cb2, cntA, cntB,
                                     ckey, cidx);
  k_sort_gather<<<BATCH, KSEL, 0, stream>>>(pcloud, ckey, cidx, out);
}